// EstDirichlet_70523363000587
// MI455X (gfx1250) — compile-verified
//
#include <hip/hip_runtime.h>

#define C 10

typedef __attribute__((ext_vector_type(2))) float v2f;
typedef __attribute__((ext_vector_type(8))) float v8f;

__global__ void EstDirichlet_zero_acc(float* __restrict__ acc) {
    if (threadIdx.x < 32) acc[threadIdx.x] = 0.0f;
}

// Phase 1: streaming softmax statistics. Grid-stride rows; per-thread register
// accumulation; per-wave cross-lane reduction via V_WMMA_F32_16X16X4_F32.
__global__ __launch_bounds__(256) void EstDirichlet_reduce(
        const float* __restrict__ x, float* __restrict__ acc, int nrows) {
    float sp[C], sp2[C], slp[C];
#pragma unroll
    for (int c = 0; c < C; ++c) { sp[c] = 0.f; sp2[c] = 0.f; slp[c] = 0.f; }

    int gid = blockIdx.x * blockDim.x + threadIdx.x;
    int stride = gridDim.x * blockDim.x;
    for (int r = gid; r < nrows; r += stride) {
        const float* row = x + (size_t)r * C;
        float v[C];
#pragma unroll
        for (int c = 0; c < C; ++c) v[c] = row[c];
        float m = v[0];
#pragma unroll
        for (int c = 1; c < C; ++c) m = fmaxf(m, v[c]);
        float e[C];
        float s = 0.f;
#pragma unroll
        for (int c = 0; c < C; ++c) { e[c] = __expf(v[c] - m); s += e[c]; }
        float inv = 1.0f / s;
        float ls  = __logf(s);
#pragma unroll
        for (int c = 0; c < C; ++c) {
            float p = e[c] * inv;
            sp[c]  += p;
            sp2[c] += p * p;
            slp[c] += (v[c] - m) - ls;   // log softmax, numerically stable
        }
    }

    // Stage partials to LDS: row = thread, col = statistic (30 used, pad to 32).
    __shared__ float T[256][33];         // stride 33 dodges bank conflicts
    int tid = threadIdx.x;
#pragma unroll
    for (int c = 0; c < C; ++c) {
        T[tid][c]      = sp[c];
        T[tid][10 + c] = sp2[c];
        T[tid][20 + c] = slp[c];
    }
    T[tid][30] = 0.f;
    T[tid][31] = 0.f;
    __syncthreads();

    // Each wave reduces its own 32x32 chunk with WMMA: D = A x ones + C.
    // A-layout (16x4 f32, ISA 7.12.2): lane<16 -> M=lane, K=0,1 ; lane>=16 -> M=lane-16, K=2,3.
    // We map A[m,k] = T[rowBase + 4*tile + k][m(+16)], so accumulating over 8
    // tiles yields D[m,*] = sum over the wave's 32 rows of column m.
    int lane     = tid & 31;
    int rowBase  = tid & ~31;
    int mrow     = lane & 15;
    int khalf    = (lane >> 4) << 1;     // 0 for lanes 0-15, 2 for lanes 16-31
    v8f acc0 = {};                       // columns 0..15  (sp[0..9], sp2[0..5])
    v8f acc1 = {};                       // columns 16..31 (sp2[6..9], slp[0..9], pad)
    v2f ones; ones[0] = 1.0f; ones[1] = 1.0f;
#pragma unroll
    for (int tile = 0; tile < 8; ++tile) {
        int k0 = rowBase + tile * 4 + khalf;
        v2f a0, a1;
        a0[0] = T[k0][mrow];      a0[1] = T[k0 + 1][mrow];
        a1[0] = T[k0][16 + mrow]; a1[1] = T[k0 + 1][16 + mrow];
        acc0 = __builtin_amdgcn_wmma_f32_16x16x4_f32(false, a0, false, ones,
                                                     (short)0, acc0, false, false);
        acc1 = __builtin_amdgcn_wmma_f32_16x16x4_f32(false, a1, false, ones,
                                                     (short)0, acc1, false, false);
    }
    // D layout: VGPR r holds row m=r for lanes 0-15, m=r+8 for lanes 16-31;
    // every column is identical, so lanes 0 and 16 cover all 16 rows of each D.
    if (lane == 0 || lane == 16) {
        int base = (lane == 0) ? 0 : 8;
#pragma unroll
        for (int rr = 0; rr < 8; ++rr) {
            atomicAdd(&acc[base + rr], acc0[rr]);
            int q = 16 + base + rr;
            if (q < 30) atomicAdd(&acc[q], acc1[rr]);
        }
    }
}

// ---- Phase 2: tiny Newton solve in fp64 (one thread) ----
__device__ double digamma_d(double x) {
    double r = 0.0;
    while (x < 6.0) { r -= 1.0 / x; x += 1.0; }
    double f = 1.0 / (x * x);
    double p = f * (1.0/12.0 - f * (1.0/120.0 - f * (1.0/252.0 - f * (1.0/240.0 - f * (1.0/132.0)))));
    return r + log(x) - 0.5 / x - p;
}

__device__ double trigamma_d(double x) {
    double r = 0.0;
    while (x < 6.0) { r += 1.0 / (x * x); x += 1.0; }
    double f = 1.0 / (x * x);
    return r + 1.0 / x + 0.5 * f
           + (f / x) * (1.0/6.0 - f * (1.0/30.0 - f * (1.0/42.0 - f * (1.0/30.0))));
}

__global__ void EstDirichlet_newton(const float* __restrict__ acc,
                                    float* __restrict__ out, int nrows) {
    if (threadIdx.x != 0 || blockIdx.x != 0) return;
    double n = (double)nrows;
    double m1[C], m2[C], lp[C], a[C];
    double ratio = 0.0;
    for (int c = 0; c < C; ++c) {
        m1[c] = (double)acc[c]      / n;
        m2[c] = (double)acc[10 + c] / n;
        lp[c] = (double)acc[20 + c] / n;
        ratio += (m1[c] - m2[c]) / (m2[c] - m1[c] * m1[c]);
    }
    ratio /= (double)C;
    for (int c = 0; c < C; ++c) a[c] = m1[c] * ratio;

    double diff = 1e30;
    for (int k = 0; k < 100 && diff >= 1e-3; ++k) {
        double asum = 0.0;
        for (int c = 0; c < C; ++c) asum += a[c];
        double dgs = digamma_d(asum);
        double z   = n * trigamma_d(asum);
        double g[C], q[C];
        double sgq = 0.0, sq = 0.0;
        for (int c = 0; c < C; ++c) {
            g[c] = (dgs - digamma_d(a[c]) + lp[c]) * n;
            q[c] = -n * trigamma_d(a[c]);
            double qi = 1.0 / q[c];
            sgq += g[c] * qi;
            sq  += qi;
        }
        double b = sgq / (1.0 / z + sq);
        diff = 0.0;
        for (int c = 0; c < C; ++c) {
            double an = a[c] - (g[c] - b) / q[c];
            diff += fabs(an - a[c]);
            a[c] = an;
        }
    }
    for (int c = 0; c < C; ++c) out[c] = (float)a[c];
}

extern "C" void kernel_launch(void* const* d_in, const int* in_sizes, int n_in,
                              void* d_out, int out_size, void* d_ws, size_t ws_size,
                              hipStream_t stream) {
    const float* x = (const float*)d_in[0];
    float* acc = (float*)d_ws;          // 32 f32 accumulators
    float* out = (float*)d_out;         // 10 f32
    int nrows = in_sizes[0] / C;

    EstDirichlet_zero_acc<<<1, 32, 0, stream>>>(acc);

    int nb = (nrows + 255) / 256;
    if (nb > 2048) nb = 2048;
    if (nb < 1) nb = 1;
    EstDirichlet_reduce<<<nb, 256, 0, stream>>>(x, acc, nrows);

    EstDirichlet_newton<<<1, 32, 0, stream>>>(acc, out, nrows);
}